// SRU_Formula_Cell_80401787781722
// MI455X (gfx1250) — compile-verified
//
#include <hip/hip_runtime.h>
#include <hip/hip_bf16.h>

// SRU forward for MI455X (gfx1250, wave32).
// Phase 1: fp32 WMMA GEMM (16x16x4) producing all 4 projections into d_ws,
//          with sigmoid/bias fused into the epilogue (projections f,r stored
//          post-sigmoid; cx stored post-bias) so the low-parallelism scan
//          keeps only the loop-carried FMA + tanh.
// Phase 2: T-sequential scan, 8192 independent (b,o) recurrences.
// Problem is HBM-bound (~0.5 GB vs 68.7 GFLOP at 23.3 TB/s) -> exact fp32
// WMMA costs nothing; grid is N-fastest so xt stays hot in the 192 MB L2.

typedef float v2f __attribute__((ext_vector_type(2)));
typedef float v8f __attribute__((ext_vector_type(8)));

#define N_IN   512
#define N_OUT  256
#define T_LEN  2048
#define BATCH  32
#define M_TOT  (T_LEN * BATCH)   // 65536
#define N_TOT  (4 * N_OUT)       // 1024

#define TM 64
#define TN 64
#define TK 64
#define LDS_STRIDE 68            // bank = (4*row + k) % 64 -> conflict-free v2f reads

__device__ __forceinline__ float sigmoidf_fast(float x) {
    return 1.0f / (1.0f + __expf(-x));
}

__global__ __launch_bounds__(128)
void sru_gemm4_kernel(const float* __restrict__ xt,
                      const float* __restrict__ Wx, const float* __restrict__ Wf,
                      const float* __restrict__ Wr, const float* __restrict__ Wc,
                      const float* __restrict__ bf, const float* __restrict__ br,
                      const float* __restrict__ bc,
                      float* __restrict__ g)
{
    // Double-buffered tiles: overlap global->LDS staging of chunk k+1 with
    // the 64 WMMAs of chunk k. 2*(64*68*4)*2 = 68 KB (<< 320 KB WGP LDS).
    __shared__ float As[2][TM * LDS_STRIDE];   // As[m][k], xt tile
    __shared__ float Bs[2][TN * LDS_STRIDE];   // Bs[n][k], weight tile

    const int tid   = threadIdx.x;
    const int wave  = tid >> 5;     // 0..3 -> M sub-strip
    const int lane  = tid & 31;
    const int lm    = lane & 15;    // A: M index / B,C: N index
    const int khalf = lane >> 4;    // selects K pair {0,1} vs {2,3}

    // N-tile on blockIdx.x (fastest): the 16 blocks sharing one A-region run
    // consecutively -> xt read from HBM ~once, weights L2-resident.
    const int nblk = blockIdx.x;            // 0..15 over N_TOT=1024
    const int m0   = blockIdx.y * TM;
    const int wsel = nblk >> 2;             // 0=Wx 1=Wf 2=Wr 3=Wc
    const int n0w  = (nblk & 3) * TN;       // row offset inside that weight
    const float* __restrict__ W =
        (wsel == 0) ? Wx : (wsel == 1) ? Wf : (wsel == 2) ? Wr : Wc;
    const float* __restrict__ bias =
        (wsel == 1) ? bf : (wsel == 2) ? br : (wsel == 3) ? bc : nullptr;

    v8f acc[4];
#pragma unroll
    for (int ns = 0; ns < 4; ++ns)
        acc[ns] = (v8f){0.f, 0.f, 0.f, 0.f, 0.f, 0.f, 0.f, 0.f};

    // ---- stage first K-chunk into buffer 0 ----
#pragma unroll
    for (int i = tid; i < TM * (TK / 4); i += 128) {
        const int r  = i >> 4;
        const int c4 = (i & 15) << 2;
        *(float4*)&As[0][r * LDS_STRIDE + c4] =
            *(const float4*)(xt + (size_t)(m0 + r) * N_IN + c4);
        *(float4*)&Bs[0][r * LDS_STRIDE + c4] =
            *(const float4*)(W + (size_t)(n0w + r) * N_IN + c4);
    }
    __syncthreads();

    int buf = 0;
    for (int kb = 0; kb < N_IN; kb += TK, buf ^= 1) {
        // ---- prefetch next K-chunk into the other buffer ----
        if (kb + TK < N_IN) {
            const int kn = kb + TK;
#pragma unroll
            for (int i = tid; i < TM * (TK / 4); i += 128) {
                const int r  = i >> 4;
                const int c4 = (i & 15) << 2;
                *(float4*)&As[buf ^ 1][r * LDS_STRIDE + c4] =
                    *(const float4*)(xt + (size_t)(m0 + r) * N_IN + kn + c4);
                *(float4*)&Bs[buf ^ 1][r * LDS_STRIDE + c4] =
                    *(const float4*)(W + (size_t)(n0w + r) * N_IN + kn + c4);
            }
        }

        // ---- 64 WMMAs on current buffer ----
        // A 16x4 fp32: lanes 0-15 hold M=lm, K={0,1}; lanes 16-31 K={2,3}.
        // B 4x16 fp32: lanes 0-15 hold N=lm, K={0,1}; lanes 16-31 K={2,3}.
        const float* Arow = &As[buf][(wave * 16 + lm) * LDS_STRIDE + 2 * khalf];
        const float* Bbase = &Bs[buf][lm * LDS_STRIDE + 2 * khalf];
#pragma unroll
        for (int kk = 0; kk < TK; kk += 4) {
            const v2f a = *(const v2f*)(Arow + kk);
#pragma unroll
            for (int ns = 0; ns < 4; ++ns) {
                const v2f b = *(const v2f*)(Bbase + ns * 16 * LDS_STRIDE + kk);
                acc[ns] = __builtin_amdgcn_wmma_f32_16x16x4_f32(
                    false, a, false, b, (short)0, acc[ns], false, false);
            }
        }
        __syncthreads();
    }

    // ---- epilogue: fuse bias + sigmoid (f, r) / bias (cx), then store ----
    // C 16x16 fp32: VGPR r -> lanes 0-15: M=r, lanes 16-31: M=r+8; N=lane%16.
    const int ncol0 = nblk * TN;
#pragma unroll
    for (int ns = 0; ns < 4; ++ns) {
        const int colw = n0w + ns * 16 + lm;     // 0..255 within this weight
        const int col  = ncol0 + ns * 16 + lm;   // 0..1023 in g
        const float bv = (wsel != 0) ? bias[colw] : 0.0f;
#pragma unroll
        for (int r = 0; r < 8; ++r) {
            const int row = m0 + wave * 16 + khalf * 8 + r;
            float v = acc[ns][r];
            if (wsel == 1 || wsel == 2)      v = sigmoidf_fast(v + bv);
            else if (wsel == 3)              v = v + bv;
            g[(size_t)row * N_TOT + col] = v;
        }
    }
}

__global__ __launch_bounds__(256)
void sru_scan_kernel(const float* __restrict__ g,
                     const float* __restrict__ ct0,
                     float* __restrict__ out)
{
    const int o = threadIdx.x;        // 0..255
    const int b = blockIdx.x;         // 0..31

    float c = ct0[b * N_OUT + o];
    const size_t colbase = (size_t)b * N_TOT + o;   // into g rows
    const size_t obase   = (size_t)b * N_OUT + o;   // into out rows
    const size_t rowstride = (size_t)BATCH * N_TOT; // 32768 floats per t

    for (int t = 0; t < T_LEN; ++t) {
        const float* row = g + (size_t)t * rowstride + colbase;

        // stream ~8 steps (1 MB) ahead on all 4 projections
        const float* pf = row + 8 * rowstride;
        __builtin_prefetch(pf + 0,         0, 0);
        __builtin_prefetch(pf + N_OUT,     0, 0);
        __builtin_prefetch(pf + 2 * N_OUT, 0, 0);
        __builtin_prefetch(pf + 3 * N_OUT, 0, 0);

        const float xp = row[0];
        const float f  = row[N_OUT];       // sigmoid applied in GEMM epilogue
        const float r  = row[2 * N_OUT];   // sigmoid applied in GEMM epilogue
        const float cx = row[3 * N_OUT];   // bias applied in GEMM epilogue

        c = f * c + (1.0f - f) * xp;       // the only loop-carried dependency
        const float h = r * tanhf(c) + (1.0f - r) * cx;

        out[(size_t)t * (BATCH * N_OUT) + obase] = h;
    }
    out[(size_t)M_TOT * N_OUT + obase] = c;   // c_final appended after ht
}

extern "C" void kernel_launch(void* const* d_in, const int* in_sizes, int n_in,
                              void* d_out, int out_size, void* d_ws, size_t ws_size,
                              hipStream_t stream) {
    const float* xt = (const float*)d_in[0];   // [2048,32,512]
    const float* ct = (const float*)d_in[1];   // [1,32,256]
    const float* Wx = (const float*)d_in[2];   // [256,512]
    const float* Wf = (const float*)d_in[3];
    const float* bf = (const float*)d_in[4];
    const float* Wr = (const float*)d_in[5];
    const float* br = (const float*)d_in[6];
    const float* Wc = (const float*)d_in[7];
    const float* bc = (const float*)d_in[8];

    float* g   = (float*)d_ws;    // needs 65536*1024*4 = 256 MiB of workspace
    float* out = (float*)d_out;   // ht [2048,32,256] ++ c_final [32,256]

    dim3 grid1(N_TOT / TN, M_TOT / TM);       // 16 x 1024, N-tile fastest
    sru_gemm4_kernel<<<grid1, 128, 0, stream>>>(xt, Wx, Wf, Wr, Wc,
                                                bf, br, bc, g);

    sru_scan_kernel<<<BATCH, N_OUT, 0, stream>>>(g, ct, out);
}